// CrossEntGroup_18769007083967
// MI455X (gfx1250) — compile-verified
//
#include <hip/hip_runtime.h>
#include <hip/hip_bf16.h>

// Problem constants (from reference): group_act [C,N,G] f32, labels [N] i32 in 0..C.
#define C_CLS 10
#define N_ROWS 500000
#define G_DIM 8

typedef __attribute__((ext_vector_type(2))) float v2f;
typedef __attribute__((ext_vector_type(8))) float v8f;

// Workspace layout (floats): [0,80) colsum[C][G], [80,720) Mpart[C][8][8], [720,730) counts (int bits)
#define WS_COLSUM 0
#define WS_MPART 80
#define WS_COUNTS 720
#define WS_TOTAL 730

__global__ void k_init(float* ws) {
    int i = blockIdx.x * blockDim.x + threadIdx.x;
    if (i < WS_TOTAL) ws[i] = 0.0f;   // int 0 and float 0.0 share the bit pattern
}

// Pass 1: per-class column sums + counts over masked rows only.
__global__ void k_colsum(const float* __restrict__ A, const int* __restrict__ labels,
                         float* __restrict__ colsum, int* __restrict__ counts) {
    __shared__ float ls[C_CLS * G_DIM];
    __shared__ int   lc[C_CLS];
    int tid = threadIdx.x;
    if (tid < C_CLS * G_DIM) ls[tid] = 0.0f;
    if (tid < C_CLS)         lc[tid] = 0;
    __syncthreads();

    int n = blockIdx.x * blockDim.x + tid;
    if (n < N_ROWS) {
        int lab = labels[n];
        if (lab > 0) {
            int c = lab - 1;
            const float4* r4 = (const float4*)(A + ((size_t)c * N_ROWS + n) * G_DIM);
            float4 lo = r4[0];
            float4 hi = r4[1];
            atomicAdd(&ls[c * G_DIM + 0], lo.x);
            atomicAdd(&ls[c * G_DIM + 1], lo.y);
            atomicAdd(&ls[c * G_DIM + 2], lo.z);
            atomicAdd(&ls[c * G_DIM + 3], lo.w);
            atomicAdd(&ls[c * G_DIM + 4], hi.x);
            atomicAdd(&ls[c * G_DIM + 5], hi.y);
            atomicAdd(&ls[c * G_DIM + 6], hi.z);
            atomicAdd(&ls[c * G_DIM + 7], hi.w);
            atomicAdd(&lc[c], 1);
        }
    }
    __syncthreads();
    if (tid < C_CLS * G_DIM) atomicAdd(&colsum[tid], ls[tid]);
    if (tid < C_CLS)         atomicAdd(&counts[tid], lc[tid]);
}

// Pass 2: per-class M[c,i,j] = sum_n p[n,i]*logp[n,j] via V_WMMA_F32_16X16X4_F32.
// One wave per block; block = (class, chunk of n).
//
// Matched row indices are queued (wave-uniform) in LDS; every 4 queued rows all 32
// lanes gather cooperatively (lane = slot*8 + g -> one 32b load each, 4 rows of
// 32B), compute p = a*(1/colsum) and log p in parallel, stage them as the K=4
// slots of one WMMA.
//
// ISA layouts (CDNA5 ISA, 32-bit 16x4 A): lanes 0-15 hold M=0..15 with
// VGPR0=K0, VGPR1=K1; lanes 16-31 hold K2/K3. B (4x16) mirrored with N on lanes.
// Rows i>=8 / cols j>=8 are zero, so D[0:8,0:8] (lanes 0-7, acc[0..7]) is the 8x8 block.
#define CHUNK 2048
#define NCHUNK ((N_ROWS + CHUNK - 1) / CHUNK)

__global__ void k_einsum_wmma(const float* __restrict__ A, const int* __restrict__ labels,
                              const float* __restrict__ colsum, float* __restrict__ Mpart) {
    __shared__ float pb[4 * G_DIM];   // staged p rows    (slot-major: pb[slot*8+g])
    __shared__ float lb[4 * G_DIM];   // staged log-p rows
    __shared__ int   q[64];           // queue of matched row indices (ring)

    const int c   = blockIdx.x / NCHUNK;
    const int n0  = (blockIdx.x % NCHUNK) * CHUNK;
    const int n1  = (n0 + CHUNK < N_ROWS) ? (n0 + CHUNK) : N_ROWS;
    const unsigned lane = threadIdx.x;           // wave32: 0..31
    const int s = (int)(lane >> 3);              // K slot this lane fills (0..3)
    const int g = (int)(lane & 7);               // group component this lane fills

    // Loop-invariant reciprocal of this class's column sum (one precise divide).
    const float rcs = 1.0f / colsum[c * G_DIM + g];

    v8f acc = {};
    int qhead = 0, qtail = 0;

    auto flush = [&]() {
        const int m  = (int)(lane & 15);         // row (A) / col (B) index
        const int kb = (lane >> 4) ? 2 : 0;      // K base for this lane half
        const int mi = (m < 8) ? m : 0;          // clamp LDS index
        const bool live = (m < 8);
        v2f a, b;
        a.x = live ? pb[(kb + 0) * G_DIM + mi] : 0.0f;
        a.y = live ? pb[(kb + 1) * G_DIM + mi] : 0.0f;
        b.x = live ? lb[(kb + 0) * G_DIM + mi] : 0.0f;
        b.y = live ? lb[(kb + 1) * G_DIM + mi] : 0.0f;
        acc = __builtin_amdgcn_wmma_f32_16x16x4_f32(false, a, false, b,
                                                    (short)0, acc, false, false);
    };

    // Gather `cnt` (1..4) queued rows in parallel, stage, and accumulate one WMMA.
    auto process = [&](int cnt) {
        const bool liveS = (s < cnt);
        float pv = 0.0f, lpv = 0.0f;
        if (liveS) {
            const int nn  = q[(qhead + s) & 63];
            const float a = A[((size_t)c * N_ROWS + nn) * G_DIM + g];
            pv  = a * rcs;
            lpv = __logf(pv);
        }
        pb[lane] = pv;                           // lane = s*8+g == slot-major layout
        lb[lane] = lpv;
        flush();
        qhead += cnt;
    };

    int lab = (n0 + (int)lane < n1) ? labels[n0 + lane] : 0;
    for (int base = n0; base < n1; base += 32) {
        // software-pipelined label load for the next ballot
        const int nn2 = base + 32 + (int)lane;
        const int nextlab = (nn2 < n1) ? labels[nn2] : 0;

        const unsigned mask = __builtin_amdgcn_ballot_w32(lab == c + 1);
        if (lab == c + 1) {
            const unsigned rank = __builtin_amdgcn_mbcnt_lo(mask, 0u);
            q[(qtail + (int)rank) & 63] = base + (int)lane;
        }
        qtail += __popc(mask);
        while (qtail - qhead >= 4) process(4);

        lab = nextlab;
    }
    if (qtail > qhead) process(qtail - qhead);   // final partial flush (zero-padded)

    // D layout: lane j (0-7), acc[i] = M[c][i][j]
    if (lane < 8) {
#pragma unroll
        for (int i = 0; i < 8; ++i)
            atomicAdd(&Mpart[(c * G_DIM + i) * G_DIM + (int)lane], acc[i]);
    }
}

// Finalize: off-diagonal sum over valid classes, normalize.
__global__ void k_final(const int* __restrict__ counts, const float* __restrict__ Mpart,
                        float* __restrict__ out) {
    __shared__ float accum;
    __shared__ int nv;
    int t = threadIdx.x;                         // 640 threads: t = ((c*8)+i)*8+j
    if (t == 0) {
        accum = 0.0f;
        int v = 0;
        for (int c = 0; c < C_CLS; ++c) v += (counts[c] >= 2) ? 1 : 0;
        nv = v;
    }
    __syncthreads();
    int c = t / 64, i = (t >> 3) & 7, j = t & 7;
    float val = ((i != j) && (counts[c] >= 2)) ? Mpart[t] : 0.0f;
    atomicAdd(&accum, val);
    __syncthreads();
    if (t == 0) out[0] = accum / ((float)nv * (float)(G_DIM * (G_DIM - 1)));
}

extern "C" void kernel_launch(void* const* d_in, const int* in_sizes, int n_in,
                              void* d_out, int out_size, void* d_ws, size_t ws_size,
                              hipStream_t stream) {
    (void)in_sizes; (void)n_in; (void)out_size; (void)ws_size;
    const float* A      = (const float*)d_in[0];
    const int*   labels = (const int*)d_in[1];
    float* ws     = (float*)d_ws;
    float* colsum = ws + WS_COLSUM;
    float* Mpart  = ws + WS_MPART;
    int*   counts = (int*)(ws + WS_COUNTS);

    k_init<<<(WS_TOTAL + 255) / 256, 256, 0, stream>>>(ws);
    k_colsum<<<(N_ROWS + 255) / 256, 256, 0, stream>>>(A, labels, colsum, counts);
    k_einsum_wmma<<<C_CLS * NCHUNK, 32, 0, stream>>>(A, labels, colsum, Mpart);
    k_final<<<1, 640, 0, stream>>>(counts, Mpart, (float*)d_out);
}